// RadarPillarFeatureNet_logits_38646115730111
// MI455X (gfx1250) — compile-verified
//
#include <hip/hip_runtime.h>
#include <math.h>

typedef float v2f __attribute__((ext_vector_type(2)));
typedef float v8f __attribute__((ext_vector_type(8)));

#define PILLAR_M 32
#define PILLAR_C 9

// ---------------------------------------------------------------------------
// Cross-lane helpers.
// dpp_add<CTRL>: x += x(dpp-permuted lane), pure VALU, no LDS.
//   0xB1 = quad_perm [1,0,3,2]  (xor1)
//   0x4E = quad_perm [2,3,0,1]  (xor2)
//   0x141 = row_half_mirror     (== xor4 once values are quad-uniform)
//   0x140 = row_mirror          (== xor8 once values are octet-uniform)
// ---------------------------------------------------------------------------
#if __has_builtin(__builtin_amdgcn_update_dpp)
template <int CTRL>
__device__ __forceinline__ float dpp_add(float x) {
    int y = __builtin_amdgcn_update_dpp(0, __float_as_int(x), CTRL, 0xF, 0xF, true);
    return x + __int_as_float(y);
}
// Input is 16-periodic across the wave (lane L == lane L^16), so xor1/2/4/8
// alone produce the full 32-point sum broadcast to every lane.
__device__ __forceinline__ float wave_sum_16periodic(float x) {
    x = dpp_add<0xB1>(x);
    x = dpp_add<0x4E>(x);
    x = dpp_add<0x141>(x);
    x = dpp_add<0x140>(x);
    return x;
}
#else
__device__ __forceinline__ float wave_sum_16periodic(float x) {
    x += __shfl_xor(x, 1, 32);
    x += __shfl_xor(x, 2, 32);
    x += __shfl_xor(x, 4, 32);
    x += __shfl_xor(x, 8, 32);
    return x;
}
#endif

// ---------------------------------------------------------------------------
// Pass 0: assemble zero-padded 16x64 weight matrix W_raw (feat-channel major),
// zero the 128 stat accumulators (sum[64], sumsq[64]).
// ---------------------------------------------------------------------------
__global__ void k_init(const float* __restrict__ w1, const float* __restrict__ w2,
                       const float* __restrict__ w3, const float* __restrict__ w4,
                       float* __restrict__ Wraw, float* __restrict__ stats) {
    int i = threadIdx.x;
    if (i < 128) stats[i] = 0.0f;
    if (i < 1024) {
        int c = i >> 6;            // feat channel 0..15
        int u = i & 63;            // output unit 0..63
        int b = u >> 4, uu = u & 15;
        int col = -1;
        if (b == 0) {              // SPATIO_IDX = [0,1,2,5,6,7,8,9]
            if (c <= 2) col = c;
            else if (c >= 5 && c <= 9) col = c - 2;
        } else if (b == 1) {       // VEL_IDX = [3,10]
            if (c == 3) col = 0; else if (c == 10) col = 1;
        } else if (b == 2) {       // SNR_IDX = [4,11]
            if (c == 4) col = 0; else if (c == 11) col = 1;
        } else {                   // LOGIT_IDX = [12,13,14,15]
            if (c >= 12) col = c - 12;
        }
        float v = 0.0f;
        if (col >= 0) {
            if (b == 0)      v = w1[uu * 8 + col];
            else if (b == 1) v = w2[uu * 2 + col];
            else if (b == 2) v = w3[uu * 2 + col];
            else             v = w4[uu * 4 + col];
        }
        Wraw[c * 64 + u] = v;
    }
}

// ---------------------------------------------------------------------------
// Per-pillar 32x16 feature build (registers only) + 32x64 GEMM via
// V_WMMA_F32_16X16X4_F32 tiles.  One wave per pillar.
//   lane L: l16 = L&15, h = L>>4.  Lane holds points p0=l16 and p1=l16+16.
//   A frag (tile t, kstep k): .x = feat[p_t][4k+2h], .y = feat[p_t][4k+2h+1]
//   B frag (ntile, k):        .x = W[(4k+2h)*64 + nt*16 + l16], .y = next row
//   D[t][nt]: lane column = nt*16 + l16, VGPR r = point 16t + r + 8h
// ---------------------------------------------------------------------------
__device__ __forceinline__ void pillar_mm(const float* __restrict__ F,
                                          const int* __restrict__ npts,
                                          const int* __restrict__ coors,
                                          int n, int l16, int h,
                                          const v2f (&Bf)[4][4],
                                          v8f (&D)[2][4]) {
    float arr[2][9];
    const float* fp = F + (size_t)n * (PILLAR_M * PILLAR_C) + l16 * PILLAR_C;
#pragma unroll
    for (int i = 0; i < 9; ++i) {
        arr[0][i] = fp[i];
        arr[1][i] = fp[16 * PILLAR_C + i];
    }
    // Sums of channels 0..4 over all 32 points.  Pair-sums are 16-periodic
    // across the wave, so a 4-step DPP butterfly yields the exact total.
    float s[5];
#pragma unroll
    for (int c = 0; c < 5; ++c) s[c] = wave_sum_16periodic(arr[0][c] + arr[1][c]);

    int np = npts[n];
    float inv = 1.0f / (float)np;                      // s[c]*inv = sum/np
    float cx = (float)coors[n * 4 + 3] * 0.2f + 0.1f;  // VX, X_OFF
    float cy = (float)coors[n * 4 + 2] * 0.2f - 39.9f; // VY, Y_OFF

    float ft[2][16];
#pragma unroll
    for (int pt = 0; pt < 2; ++pt) {
        float mk = ((l16 + 16 * pt) < np) ? 1.0f : 0.0f;
        float fc0 = arr[pt][0] - cx;
        float fc1 = arr[pt][1] - cy;
        ft[pt][0] = fc0 * mk;
        ft[pt][1] = fc1 * mk;
#pragma unroll
        for (int c = 2; c < 9; ++c) ft[pt][c] = arr[pt][c] * mk;
        ft[pt][9]  = (arr[pt][0] - s[0] * inv) * mk;   // f_cluster x
        ft[pt][10] = (arr[pt][1] - s[1] * inv) * mk;   // f_cluster y
        ft[pt][11] = (arr[pt][2] - s[2] * inv) * mk;   // f_cluster z
        ft[pt][12] = ft[pt][0];
        ft[pt][13] = ft[pt][1];
        ft[pt][14] = (arr[pt][3] - s[3] * inv) * mk;   // vs_center x
        ft[pt][15] = (arr[pt][4] - s[4] * inv) * mk;   // vs_center y
    }

    v2f A[2][4];
#pragma unroll
    for (int t = 0; t < 2; ++t) {
#pragma unroll
        for (int k = 0; k < 4; ++k) {
            A[t][k].x = h ? ft[t][4 * k + 2] : ft[t][4 * k + 0];
            A[t][k].y = h ? ft[t][4 * k + 3] : ft[t][4 * k + 1];
        }
    }
#pragma unroll
    for (int t = 0; t < 2; ++t) {
#pragma unroll
        for (int nt = 0; nt < 4; ++nt) {
            v8f acc = {};
#pragma unroll
            for (int k = 0; k < 4; ++k) {
                acc = __builtin_amdgcn_wmma_f32_16x16x4_f32(
                    false, A[t][k], false, Bf[nt][k], (short)0, acc, false, false);
            }
            D[t][nt] = acc;
        }
    }
}

__device__ __forceinline__ void load_bfrag(const float* __restrict__ W,
                                           int l16, int h, v2f (&Bf)[4][4]) {
#pragma unroll
    for (int nt = 0; nt < 4; ++nt) {
#pragma unroll
        for (int k = 0; k < 4; ++k) {
            Bf[nt][k].x = W[(4 * k + 2 * h) * 64 + nt * 16 + l16];
            Bf[nt][k].y = W[(4 * k + 2 * h + 1) * 64 + nt * 16 + l16];
        }
    }
}

// ---------------------------------------------------------------------------
// Pass 1: streaming stats.  Per-unit sum(y), sum(y^2) over all N*M rows
// (masked rows contribute exact zeros, matching the reference mean/var).
// ---------------------------------------------------------------------------
__global__ void k_stats(const float* __restrict__ F, const int* __restrict__ npts,
                        const int* __restrict__ coors, const float* __restrict__ W,
                        float* __restrict__ stats, int N) {
    int lane = threadIdx.x & 31;
    int l16 = lane & 15, h = lane >> 4;
    int wid = (blockIdx.x * blockDim.x + threadIdx.x) >> 5;
    int nw  = (gridDim.x * blockDim.x) >> 5;

    v2f Bf[4][4];
    load_bfrag(W, l16, h, Bf);

    float sAcc[4] = {0.f, 0.f, 0.f, 0.f};
    float qAcc[4] = {0.f, 0.f, 0.f, 0.f};

    for (int n = wid; n < N; n += nw) {
        v8f D[2][4];
        pillar_mm(F, npts, coors, n, l16, h, Bf, D);
#pragma unroll
        for (int nt = 0; nt < 4; ++nt) {
#pragma unroll
            for (int t = 0; t < 2; ++t) {
#pragma unroll
                for (int r = 0; r < 8; ++r) {
                    float v = D[t][nt][r];
                    sAcc[nt] += v;
                    qAcc[nt] += v * v;
                }
            }
        }
    }
#pragma unroll
    for (int nt = 0; nt < 4; ++nt) {
        sAcc[nt] += __shfl_xor(sAcc[nt], 16, 32);
        qAcc[nt] += __shfl_xor(qAcc[nt], 16, 32);
    }
    if (lane < 16) {
#pragma unroll
        for (int nt = 0; nt < 4; ++nt) {
            atomicAdd(&stats[nt * 16 + lane], sAcc[nt]);
            atomicAdd(&stats[64 + nt * 16 + lane], qAcc[nt]);
        }
    }
}

// ---------------------------------------------------------------------------
// Pass 2: fold BN into effective weights + bias.
//   y_bn = (y - mu)*rsqrt(var+eps)*g + b  =  feat @ (W*a) + (b - mu*a)
// ---------------------------------------------------------------------------
__global__ void k_fold(const float* __restrict__ Wraw, const float* __restrict__ stats,
                       const float* __restrict__ gamma, const float* __restrict__ beta,
                       float* __restrict__ Weff, float* __restrict__ biasv, float NM) {
    int u = threadIdx.x;
    if (u >= 64) return;
    int b = u >> 4, uu = u & 15;
    float invNM = 1.0f / NM;
    float mu  = stats[u] * invNM;
    float var = stats[64 + u] * invNM - mu * mu;
    float a = gamma[b * 16 + uu] / sqrtf(var + 1e-3f);
    biasv[u] = beta[b * 16 + uu] - mu * a;
#pragma unroll
    for (int c = 0; c < 16; ++c) Weff[c * 64 + u] = Wraw[c * 64 + u] * a;
}

// ---------------------------------------------------------------------------
// Pass 3: recompute y with folded weights, +bias, max over 32 points, ReLU.
// Masked rows give y = bias_eff, which correctly participates in the max.
// ---------------------------------------------------------------------------
__global__ void k_out(const float* __restrict__ F, const int* __restrict__ npts,
                      const int* __restrict__ coors, const float* __restrict__ W,
                      const float* __restrict__ biasv, float* __restrict__ out, int N) {
    int lane = threadIdx.x & 31;
    int l16 = lane & 15, h = lane >> 4;
    int wid = (blockIdx.x * blockDim.x + threadIdx.x) >> 5;
    int nw  = (gridDim.x * blockDim.x) >> 5;

    v2f Bf[4][4];
    load_bfrag(W, l16, h, Bf);
    float bu[4];
#pragma unroll
    for (int nt = 0; nt < 4; ++nt) bu[nt] = biasv[nt * 16 + l16];

    for (int n = wid; n < N; n += nw) {
        v8f D[2][4];
        pillar_mm(F, npts, coors, n, l16, h, Bf, D);
#pragma unroll
        for (int nt = 0; nt < 4; ++nt) {
            float m = D[0][nt][0];
#pragma unroll
            for (int t = 0; t < 2; ++t) {
#pragma unroll
                for (int r = 0; r < 8; ++r) m = fmaxf(m, D[t][nt][r]);
            }
            m += bu[nt];                             // max(y)+b == max(y+b)
            m = fmaxf(m, __shfl_xor(m, 16, 32));     // fold row halves
            m = fmaxf(m, 0.0f);                      // ReLU (monotone w/ max)
            if (lane < 16) out[(size_t)n * 64 + nt * 16 + lane] = m;
        }
    }
}

// ---------------------------------------------------------------------------
// inputs: 0 features(N,32,9) f32, 1..4 w1..w4 f32, 5 gamma(4,16), 6 beta(4,16),
//         7 num_points(N) i32, 8 coors(N,4) i32.  out: (N,64) f32.
// ws: [0,4KB) W_raw | [4KB,8KB) W_eff | [8KB,+256B) bias | [8448,+512B) stats
// ---------------------------------------------------------------------------
extern "C" void kernel_launch(void* const* d_in, const int* in_sizes, int n_in,
                              void* d_out, int out_size, void* d_ws, size_t ws_size,
                              hipStream_t stream) {
    const float* F    = (const float*)d_in[0];
    const float* w1   = (const float*)d_in[1];
    const float* w2   = (const float*)d_in[2];
    const float* w3   = (const float*)d_in[3];
    const float* w4   = (const float*)d_in[4];
    const float* gm   = (const float*)d_in[5];
    const float* bt   = (const float*)d_in[6];
    const int* npts   = (const int*)d_in[7];
    const int* coors  = (const int*)d_in[8];
    float* out = (float*)d_out;

    char* ws = (char*)d_ws;
    float* Wraw  = (float*)(ws);
    float* Weff  = (float*)(ws + 4096);
    float* biasv = (float*)(ws + 8192);
    float* stats = (float*)(ws + 8448);

    int N = in_sizes[7];

    k_init<<<1, 1024, 0, stream>>>(w1, w2, w3, w4, Wraw, stats);
    k_stats<<<1024, 256, 0, stream>>>(F, npts, coors, Wraw, stats, N);
    k_fold<<<1, 64, 0, stream>>>(Wraw, stats, gm, bt, Weff, biasv, (float)N * 32.0f);
    k_out<<<1024, 256, 0, stream>>>(F, npts, coors, Weff, biasv, out, N);
}